// TemporalRoIPool_76605036691592
// MI455X (gfx1250) — compile-verified
//
#include <hip/hip_runtime.h>
#include <cstdint>
#include <cstddef>

// TemporalRoIPool on MI455X (gfx1250): bandwidth-bound 2-tap temporal lerp.
// Strategy: per-(b,q) block, per-sample wave; gathered feature rows are moved
// with CDNA5 async global->LDS b128 copies (ASYNCcnt) and drained with split
// s_wait_asynccnt waits so compute overlaps the remaining in-flight copies.

namespace {

constexpr int kT  = 2048;   // time steps
constexpr int kD  = 512;    // feature dim
constexpr int kS  = 8;      // OUTPUT_SIZE samples per span
constexpr int kNq = 300;    // queries per batch

// Builtin parameter types (from hipcc diagnostic):
//   arg0: int __vector(4) __device__ *   -> AS(1) pointer to 16B int vector
//   arg1: AS(3) pointer to 16B int vector
typedef int v4i __attribute__((vector_size(16)));
typedef __attribute__((address_space(1))) v4i* gv4i_p;
typedef __attribute__((address_space(3))) v4i* lv4i_p;

__device__ __forceinline__ void async_load_b128_to_lds(const float* gsrc,
                                                       float* lds_dst) {
#if defined(__HIP_DEVICE_COMPILE__)
#if __has_builtin(__builtin_amdgcn_global_load_async_to_lds_b128)
  __builtin_amdgcn_global_load_async_to_lds_b128((gv4i_p)(void*)gsrc,
                                                 (lv4i_p)(void*)lds_dst,
                                                 /*offset=*/0, /*cpol=*/0);
#else
  unsigned lofs = (unsigned)(unsigned long long)(lv4i_p)(void*)lds_dst;
  asm volatile("global_load_async_to_lds_b128 %0, %1, off"
               :: "v"(lofs), "v"(gsrc)
               : "memory");
#endif
#endif
}

template <int N>
__device__ __forceinline__ void wait_asynccnt_le() {
#if defined(__HIP_DEVICE_COMPILE__)
#if __has_builtin(__builtin_amdgcn_s_wait_asynccnt)
  __builtin_amdgcn_s_wait_asynccnt(N);
#else
  asm volatile("s_wait_asynccnt %0" :: "n"(N) : "memory");
#endif
#endif
}

}  // namespace

__global__ __launch_bounds__(256, 1)
void temporal_roi_sample_kernel(const float* __restrict__ vf,
                                const float* __restrict__ spans,
                                float* __restrict__ out) {
  const int bq   = blockIdx.x;         // b * kNq + q
  const int wave = threadIdx.x >> 5;   // sample index s in [0, 8)
  const int lane = threadIdx.x & 31;
  const int b    = bq / kNq;

  // Block-uniform span -> scalar loads.
  const float s0 = spans[2 * bq + 0];
  const float s1 = spans[2 * bq + 1];
  const float start = s0 * (float)(kT - 1);
  const float endv  = s1 * (float)(kT - 1);
  const float p = start + ((float)wave * (1.0f / 7.0f)) * (endv - start);

  // p >= 0, so int-cast truncation == floor (matches reference .astype(int32)).
  int idx_f = (int)p;
  idx_f = idx_f < 0 ? 0 : (idx_f > kT - 2 ? kT - 2 : idx_f);
  const int idx_c = idx_f + 1;                 // always <= kT-1
  const float wc = p - (float)idx_f;
  const float wf = 1.0f - wc;

  const float* rowf = vf + ((size_t)b * kT + idx_f) * kD;
  const float* rowc = vf + ((size_t)b * kT + idx_c) * kD;

  // 8 waves * 2 rows * 512 f32 = 32 KiB (<< 320 KiB/WGP).
  __shared__ __align__(16) float smem[kS * 2 * kD];
  float* lrow = smem + wave * (2 * kD);

  // Issue interleaved async copies: (floor, ceil) per 128-float chunk.
  // Each instruction moves a coalesced 512 B per wave straight into LDS.
#pragma unroll
  for (int c = 0; c < 4; ++c) {
    const int col = c * 128 + lane * 4;
    async_load_b128_to_lds(rowf + col, lrow + col);
    async_load_b128_to_lds(rowc + col, lrow + kD + col);
  }

  float* outp = out + ((size_t)bq * kS + wave) * kD;

  auto chunk = [&](int c) {
    const int col = c * 128 + lane * 4;
    const float4 ff = *(const float4*)(lrow + col);
    const float4 fc = *(const float4*)(lrow + kD + col);
    float4 r;
    r.x = wf * ff.x + wc * fc.x;
    r.y = wf * ff.y + wc * fc.y;
    r.z = wf * ff.z + wc * fc.z;
    r.w = wf * ff.w + wc * fc.w;
    *(float4*)(outp + col) = r;
  };

  // Async loads retire in issue order: after 8 issued, ASYNCcnt<=6 means the
  // first (floor,ceil) pair of chunk 0 landed in LDS, etc. Split waits let the
  // lerp+store of chunk k overlap the copies of chunks k+1..3.
  wait_asynccnt_le<6>(); chunk(0);
  wait_asynccnt_le<4>(); chunk(1);
  wait_asynccnt_le<2>(); chunk(2);
  wait_asynccnt_le<0>(); chunk(3);
}

extern "C" void kernel_launch(void* const* d_in, const int* in_sizes, int n_in,
                              void* d_out, int out_size, void* d_ws, size_t ws_size,
                              hipStream_t stream) {
  const float* vf    = (const float*)d_in[0];   // [16, 2048, 512] f32
  const float* spans = (const float*)d_in[1];   // [16, 300, 2]  f32
  float* out = (float*)d_out;                   // [16, 300, 8, 512] f32

  const int bq_count = in_sizes[1] / 2;         // B * Nq = 4800 blocks
  dim3 grid(bq_count), block(256);
  hipLaunchKernelGGL(temporal_roi_sample_kernel, grid, block, 0, stream,
                     vf, spans, out);
}